// PREDICT_EMISSION_3D_GRID_48155173323452
// MI455X (gfx1250) — compile-verified
//
#include <hip/hip_runtime.h>
#include <hip/hip_bf16.h>
#include <stdint.h>

// PREDICT_EMISSION_3D_GRID: Rodrigues velocity-warp + trilinear grid sample.
// x,y,z: [4096*8192] f32; t,v: [1] f32; axis: [3] f32; grid: [64^3] f32.
// out: emission [4096*8192] f32.
//
// CDNA5 data path: GLOBAL_LOAD_ASYNC_TO_LDS_B128 double-buffered staging of the
// x/y/z streams (ASYNCcnt-tracked, s_wait_asynccnt), ds_load_b128 consumption,
// global_prefetch_b8 to warm L2 ahead of the async window.

#define GRID_RES 64
#define GR2      (GRID_RES * GRID_RES)
#define F32_MAX  3.402823466e38f
#define TPB      256   // threads per block
#define NT       8     // tiles of TPB quads per block

// ---- trilinear sample of 64^3 grid, JAX map_coordinates(order=1, cval=0) ----
__device__ __forceinline__ float trilerp64(const float* __restrict__ g,
                                           float c0, float c1, float c2) {
  float f0 = floorf(c0), f1 = floorf(c1), f2 = floorf(c2);
  int i0 = (int)f0, i1 = (int)f1, i2 = (int)f2;
  float t0 = c0 - f0, t1 = c1 - f1, t2 = c2 - f2;

  float w0a = ((unsigned)i0       < 64u) ? (1.0f - t0) : 0.0f;
  float w0b = ((unsigned)(i0 + 1) < 64u) ? t0          : 0.0f;
  float w1a = ((unsigned)i1       < 64u) ? (1.0f - t1) : 0.0f;
  float w1b = ((unsigned)(i1 + 1) < 64u) ? t1          : 0.0f;
  float w2a = ((unsigned)i2       < 64u) ? (1.0f - t2) : 0.0f;
  float w2b = ((unsigned)(i2 + 1) < 64u) ? t2          : 0.0f;

  int a0 = min(max(i0, 0), 63), b0 = min(max(i0 + 1, 0), 63);
  int a1 = min(max(i1, 0), 63), b1 = min(max(i1 + 1, 0), 63);
  int a2 = min(max(i2, 0), 63), b2 = min(max(i2 + 1, 0), 63);

  const float* ra = g + a0 * GR2;
  const float* rb = g + b0 * GR2;
  int ja = a1 * GRID_RES, jb = b1 * GRID_RES;

  float g000 = ra[ja + a2], g001 = ra[ja + b2];
  float g010 = ra[jb + a2], g011 = ra[jb + b2];
  float g100 = rb[ja + a2], g101 = rb[ja + b2];
  float g110 = rb[jb + a2], g111 = rb[jb + b2];

  float e0 = w1a * (w2a * g000 + w2b * g001) + w1b * (w2a * g010 + w2b * g011);
  float e1 = w1a * (w2a * g100 + w2b * g101) + w1b * (w2a * g110 + w2b * g111);
  return w0a * e0 + w0b * e1;
}

// ---- one point: Rodrigues rotate, mask non-finite, sample grid ----
__device__ __forceinline__ float warp_sample(float px, float py, float pz,
                                             float ntv,  // -t*v
                                             float kx, float ky, float kz,
                                             const float* __restrict__ g) {
  float r2 = fmaf(px, px, fmaf(py, py, pz * pz));
  float ir = rsqrtf(r2);       // r^-1   (r2==0 -> +inf)
  float w15 = ir * sqrtf(ir);  // r^-1.5
  float theta = ntv * w15;     // inf -> sin/cos -> NaN -> masked below
  float s = __sinf(theta);
  float c = __cosf(theta);

  float xx = ky * pz - kz * py;
  float xy = kz * px - kx * pz;
  float xz = kx * py - ky * px;
  float kdp = kx * px + ky * py + kz * pz;
  float omc = 1.0f - c;

  float ox = px * c + xx * s + kx * kdp * omc;
  float oy = py * c + xy * s + ky * kdp * omc;
  float oz = pz * c + xz * s + kz * kdp * omc;

  bool v0 = (fabsf(ox) <= F32_MAX);
  ox = v0 ? ox : 0.0f;
  oy = (fabsf(oy) <= F32_MAX) ? oy : 0.0f;
  oz = (fabsf(oz) <= F32_MAX) ? oz : 0.0f;

  // (p + FOV/2) / FOV * (GRID_RES-1)  ==  p*6.3 + 31.5
  float c0 = fmaf(ox, 6.3f, 31.5f);
  float c1 = fmaf(oy, 6.3f, 31.5f);
  float c2 = fmaf(oz, 6.3f, 31.5f);

  float e = trilerp64(g, c0, c1, c2);
  return v0 ? e : 0.0f;
}

// ---- CDNA5 async global->LDS copy of one 16B element (per lane) ----
__device__ __forceinline__ void async_stage16(uint32_t lds_off, const void* gaddr) {
  asm volatile("global_load_async_to_lds_b128 %0, %1, off"
               :: "v"(lds_off), "v"((uint64_t)(uintptr_t)gaddr)
               : "memory");
}

// ---- main kernel: double-buffered async LDS staging, 4 points/thread ----
__global__ __launch_bounds__(TPB) void predict_emission_async(
    const float4* __restrict__ X, const float4* __restrict__ Y,
    const float4* __restrict__ Z,
    const float* __restrict__ T, const float* __restrict__ V,
    const float* __restrict__ A, const float* __restrict__ G,
    float4* __restrict__ O, int n4) {
  __shared__ float4 buf[2][3][TPB];  // 24 KB: 2 tiles x {x,y,z} x 256 quads

  const int tid = threadIdx.x;
  const long long base = (long long)blockIdx.x * (NT * TPB);

  // wave-uniform scalars (compiler emits s_load for these)
  float ntv = -T[0] * V[0];
  float ax = A[0], ay = A[1], az = A[2];
  float kn = rsqrtf(ax * ax + ay * ay + az * az);
  float kx = ax * kn, ky = ay * kn, kz = az * kn;

  auto issue = [&](int j, int b) {
    long long q = base + (long long)j * TPB + tid;
    long long qc = q < (long long)n4 - 1 ? q : (long long)n4 - 1;  // clamp: safe dup loads
    if (qc < 0) qc = 0;
    uint32_t lx = (uint32_t)(uintptr_t)&buf[b][0][tid];
    uint32_t ly = (uint32_t)(uintptr_t)&buf[b][1][tid];
    uint32_t lz = (uint32_t)(uintptr_t)&buf[b][2][tid];
    // warm L2 one block-chunk ahead of the async window
    __builtin_prefetch(X + qc + NT * TPB, 0, 3);
    async_stage16(lx, X + qc);
    async_stage16(ly, Y + qc);
    async_stage16(lz, Z + qc);
  };

  issue(0, 0);

  for (int j = 0; j < NT; ++j) {
    int b = j & 1;
    if (j + 1 < NT) {
      issue(j + 1, (j + 1) & 1);
      // 3 loads of tile j+1 may remain in flight; tile j complete (in-order)
      asm volatile("s_wait_asynccnt 0x3" ::: "memory");
    } else {
      asm volatile("s_wait_asynccnt 0x0" ::: "memory");
    }

    float4 xv = buf[b][0][tid];  // ds_load_b128
    float4 yv = buf[b][1][tid];
    float4 zv = buf[b][2][tid];

    float px[4] = {xv.x, xv.y, xv.z, xv.w};
    float py[4] = {yv.x, yv.y, yv.z, yv.w};
    float pz[4] = {zv.x, zv.y, zv.z, zv.w};
    float e[4];
#pragma unroll
    for (int u = 0; u < 4; ++u)
      e[u] = warp_sample(px[u], py[u], pz[u], ntv, kx, ky, kz, G);

    long long q = base + (long long)j * TPB + tid;
    if (q < (long long)n4)
      O[q] = make_float4(e[0], e[1], e[2], e[3]);
  }
}

// ---- scalar tail (n % 4 leftovers; not hit for 4096*8192, kept generic) ----
__global__ void predict_emission_tail(
    const float* __restrict__ x, const float* __restrict__ y,
    const float* __restrict__ z,
    const float* __restrict__ T, const float* __restrict__ V,
    const float* __restrict__ A, const float* __restrict__ G,
    float* __restrict__ out, int start, int n) {
  int i = start + (int)(blockIdx.x * blockDim.x + threadIdx.x);
  if (i >= n) return;
  float ntv = -T[0] * V[0];
  float ax = A[0], ay = A[1], az = A[2];
  float kn = rsqrtf(ax * ax + ay * ay + az * az);
  out[i] = warp_sample(x[i], y[i], z[i], ntv, ax * kn, ay * kn, az * kn, G);
}

extern "C" void kernel_launch(void* const* d_in, const int* in_sizes, int n_in,
                              void* d_out, int out_size, void* d_ws, size_t ws_size,
                              hipStream_t stream) {
  (void)n_in; (void)d_ws; (void)ws_size; (void)out_size;
  const float* x = (const float*)d_in[0];
  const float* y = (const float*)d_in[1];
  const float* z = (const float*)d_in[2];
  const float* t = (const float*)d_in[3];
  const float* v = (const float*)d_in[4];
  const float* a = (const float*)d_in[5];
  const float* g = (const float*)d_in[6];
  float* out = (float*)d_out;

  int n = in_sizes[0];
  int n4 = n >> 2;
  int rem = n & 3;

  if (n4 > 0) {
    int quads_per_block = NT * TPB;
    int blocks = (n4 + quads_per_block - 1) / quads_per_block;
    predict_emission_async<<<blocks, TPB, 0, stream>>>(
        (const float4*)x, (const float4*)y, (const float4*)z,
        t, v, a, g, (float4*)out, n4);
  }
  if (rem > 0) {
    predict_emission_tail<<<1, 32, 0, stream>>>(x, y, z, t, v, a, g, out,
                                                n4 << 2, n);
  }
}